// Critic_63230508531831
// MI455X (gfx1250) — compile-verified
//
#include <hip/hip_runtime.h>
#include <hip/hip_bf16.h>

typedef float v2f __attribute__((ext_vector_type(2)));
typedef float v8f __attribute__((ext_vector_type(8)));

#define H1D 40
#define G1D 24
#define H2D 24
#define G2D 24
#define ADIMD 8

// ---------------------------------------------------------------------------
// Degree counting + normalization
// ---------------------------------------------------------------------------
__global__ __launch_bounds__(256) void deg_count_kernel(
    const int* __restrict__ src, const int* __restrict__ dst,
    int* __restrict__ odeg, int* __restrict__ ideg, int E)
{
  int i = blockIdx.x * blockDim.x + threadIdx.x;
  if (i >= E) return;
  atomicAdd(&odeg[src[i]], 1);
  atomicAdd(&ideg[dst[i]], 1);
}

__global__ __launch_bounds__(256) void deg_norm_kernel(
    const int* __restrict__ odeg, const int* __restrict__ ideg,
    float* __restrict__ cs, float* __restrict__ cd, int N)
{
  int i = blockIdx.x * blockDim.x + threadIdx.x;
  if (i >= N) return;
  cs[i] = rsqrtf(fmaxf((float)odeg[i], 1.0f));
  cd[i] = rsqrtf(fmaxf((float)ideg[i], 1.0f));
}

// ---------------------------------------------------------------------------
// FC1: h1 = relu([n_delay|n_res] @ fc1_w^T + fc1_b)   [N,16]x[16,40]
// One 16-node tile per wave32; WMMA f32 16x16x4, K=16 (4 steps), 3 col tiles.
// ---------------------------------------------------------------------------
__device__ __forceinline__ float ip_elem(const float* nd, const float* nr, int node, int k) {
  return (k < 8) ? nd[node * 8 + k] : nr[node * 8 + (k - 8)];
}

__global__ __launch_bounds__(256) void fc1_wmma_kernel(
    const float* __restrict__ nd, const float* __restrict__ nr,
    const float* __restrict__ w /*[40,16]*/, const float* __restrict__ bias,
    float* __restrict__ h1, int N)
{
  const int lane = threadIdx.x & 31;
  const int wave = threadIdx.x >> 5;
  const int ntiles = (N + 15) >> 4;
  const int tile = blockIdx.x * 8 + wave;
  if (tile >= ntiles) return;                 // wave-uniform: EXEC stays all-1s
  const int n = lane & 15, hi = lane >> 4;
  const bool full = (tile * 16 + 16) <= N;    // wave-uniform fast path
  int node = tile * 16 + n;
  int rnode = node < N ? node : N - 1;

  v2f afrag[4];
#pragma unroll
  for (int s = 0; s < 4; ++s) {
    int k = s * 4 + 2 * hi;
    afrag[s][0] = ip_elem(nd, nr, rnode, k);
    afrag[s][1] = ip_elem(nd, nr, rnode, k + 1);
  }
#pragma unroll
  for (int j0 = 0; j0 < 48; j0 += 16) {
    int j = j0 + n;
    v8f acc = {};
#pragma unroll
    for (int s = 0; s < 4; ++s) {
      int k = s * 4 + 2 * hi;
      v2f b;
      b[0] = (j < H1D) ? w[j * 16 + k]     : 0.0f;
      b[1] = (j < H1D) ? w[j * 16 + k + 1] : 0.0f;
      acc = __builtin_amdgcn_wmma_f32_16x16x4_f32(false, afrag[s], false, b,
                                                  (short)0, acc, false, false);
    }
    if (j < H1D) {
      float bj = bias[j];
      float* p = h1 + (size_t)(tile * 16 + 8 * hi) * H1D + j;
      if (full) {
#pragma unroll
        for (int v = 0; v < 8; ++v)
          p[(size_t)v * H1D] = fmaxf(acc[v] + bj, 0.0f);
      } else {
#pragma unroll
        for (int v = 0; v < 8; ++v) {
          int m = tile * 16 + v + 8 * hi;
          if (m < N) h1[(size_t)m * H1D + j] = fmaxf(acc[v] + bj, 0.0f);
        }
      }
    }
  }
}

// ---------------------------------------------------------------------------
// BN statistics: per-column sum and sum-of-squares over all N nodes
// ---------------------------------------------------------------------------
__global__ __launch_bounds__(256) void bn_stats_kernel(
    const float* __restrict__ h1, float* __restrict__ sums, int N)
{
  __shared__ float ls[2 * H1D];
  if (threadIdx.x < 2 * H1D) ls[threadIdx.x] = 0.0f;
  __syncthreads();
  size_t total = (size_t)N * H1D;
  size_t stride = (size_t)gridDim.x * blockDim.x;
  for (size_t idx = blockIdx.x * (size_t)blockDim.x + threadIdx.x; idx < total; idx += stride) {
    int c = (int)(idx % H1D);
    float v = h1[idx];
    atomicAdd(&ls[c], v);
    atomicAdd(&ls[H1D + c], v * v);
  }
  __syncthreads();
  if (threadIdx.x < 2 * H1D) atomicAdd(&sums[threadIdx.x], ls[threadIdx.x]);
}

// stat[c] = A = g*rsqrt(var+eps); stat[40+c] = B = b - mu*A   (h1n = h1*A + B)
__global__ void bn_finalize_kernel(
    const float* __restrict__ sums, const float* __restrict__ g,
    const float* __restrict__ b, float* __restrict__ stat, float invN)
{
  int c = threadIdx.x;
  if (c >= H1D) return;
  float mu  = sums[c] * invN;
  float var = sums[H1D + c] * invN - mu * mu;
  float A   = g[c] * rsqrtf(var + 1e-5f);
  stat[c]        = A;
  stat[H1D + c]  = b[c] - mu * A;
}

// ---------------------------------------------------------------------------
// Conv1 pre: h1w = cs * ((h1*A + B) @ conv1_w)   [N,40]x[40,24]
// ---------------------------------------------------------------------------
__global__ __launch_bounds__(256) void conv1_gemm_kernel(
    const float* __restrict__ h1, const float* __restrict__ stat,
    const float* __restrict__ w /*[40,24]*/, const float* __restrict__ cs,
    float* __restrict__ out, int N)
{
  const int lane = threadIdx.x & 31;
  const int wave = threadIdx.x >> 5;
  const int ntiles = (N + 15) >> 4;
  const int tile = blockIdx.x * 8 + wave;
  if (tile >= ntiles) return;
  const int n = lane & 15, hi = lane >> 4;
  const bool full = (tile * 16 + 16) <= N;
  int node = tile * 16 + n;
  int rnode = node < N ? node : N - 1;

  v2f afrag[10];
#pragma unroll
  for (int s = 0; s < 10; ++s) {
    int k = s * 4 + 2 * hi;
    afrag[s][0] = h1[(size_t)rnode * H1D + k]     * stat[k]     + stat[H1D + k];
    afrag[s][1] = h1[(size_t)rnode * H1D + k + 1] * stat[k + 1] + stat[H1D + k + 1];
  }
  float csr[8];
#pragma unroll
  for (int v = 0; v < 8; ++v) {
    int m = tile * 16 + v + 8 * hi;
    csr[v] = cs[m < N ? m : N - 1];
  }
#pragma unroll
  for (int j0 = 0; j0 < 32; j0 += 16) {
    int j = j0 + n;
    v8f acc = {};
#pragma unroll
    for (int s = 0; s < 10; ++s) {
      int k = s * 4 + 2 * hi;
      v2f b;
      b[0] = (j < G1D) ? w[k * G1D + j]       : 0.0f;
      b[1] = (j < G1D) ? w[(k + 1) * G1D + j] : 0.0f;
      acc = __builtin_amdgcn_wmma_f32_16x16x4_f32(false, afrag[s], false, b,
                                                  (short)0, acc, false, false);
    }
    if (j < G1D) {
      float* p = out + (size_t)(tile * 16 + 8 * hi) * G1D + j;
      if (full) {
#pragma unroll
        for (int v = 0; v < 8; ++v)
          p[(size_t)v * G1D] = acc[v] * csr[v];
      } else {
#pragma unroll
        for (int v = 0; v < 8; ++v) {
          int m = tile * 16 + v + 8 * hi;
          if (m < N) out[(size_t)m * G1D + j] = acc[v] * csr[v];
        }
      }
    }
  }
}

// ---------------------------------------------------------------------------
// Edge scatter: agg[dst] += h[src]  (flat over E*24, float atomics)
// ---------------------------------------------------------------------------
__global__ __launch_bounds__(256) void scatter_kernel(
    const float* __restrict__ h, const int* __restrict__ src,
    const int* __restrict__ dst, float* __restrict__ agg, unsigned total)
{
  unsigned idx = blockIdx.x * 256u + threadIdx.x;
  if (idx >= total) return;
  unsigned e = idx / 24u;
  unsigned c = idx - e * 24u;
  float v = h[(size_t)src[e] * 24 + c];
  atomicAdd(&agg[(size_t)dst[e] * 24 + c], v);
}

// ---------------------------------------------------------------------------
// FC2: h2 = relu([agg1*cd + c1b | action] @ fc2_w^T + fc2_b)  [N,32]x[32,24]
// ---------------------------------------------------------------------------
__global__ __launch_bounds__(256) void fc2_wmma_kernel(
    const float* __restrict__ agg1, const float* __restrict__ cd,
    const float* __restrict__ c1b, const float* __restrict__ act,
    const float* __restrict__ w /*[24,32]*/, const float* __restrict__ bias,
    float* __restrict__ h2, int N)
{
  const int lane = threadIdx.x & 31;
  const int wave = threadIdx.x >> 5;
  const int ntiles = (N + 15) >> 4;
  const int tile = blockIdx.x * 8 + wave;
  if (tile >= ntiles) return;
  const int n = lane & 15, hi = lane >> 4;
  const bool full = (tile * 16 + 16) <= N;
  int node = tile * 16 + n;
  int rnode = node < N ? node : N - 1;
  float cdn = cd[rnode];

  v2f afrag[8];
#pragma unroll
  for (int s = 0; s < 8; ++s) {
    int k = s * 4 + 2 * hi;
    afrag[s][0] = (k < G1D)  ? agg1[(size_t)rnode * G1D + k] * cdn + c1b[k]
                             : act[(size_t)rnode * ADIMD + (k - G1D)];
    int k1 = k + 1;
    afrag[s][1] = (k1 < G1D) ? agg1[(size_t)rnode * G1D + k1] * cdn + c1b[k1]
                             : act[(size_t)rnode * ADIMD + (k1 - G1D)];
  }
#pragma unroll
  for (int j0 = 0; j0 < 32; j0 += 16) {
    int j = j0 + n;
    v8f acc = {};
#pragma unroll
    for (int s = 0; s < 8; ++s) {
      int k = s * 4 + 2 * hi;
      v2f b;
      b[0] = (j < H2D) ? w[j * 32 + k]     : 0.0f;
      b[1] = (j < H2D) ? w[j * 32 + k + 1] : 0.0f;
      acc = __builtin_amdgcn_wmma_f32_16x16x4_f32(false, afrag[s], false, b,
                                                  (short)0, acc, false, false);
    }
    if (j < H2D) {
      float bj = bias[j];
      float* p = h2 + (size_t)(tile * 16 + 8 * hi) * H2D + j;
      if (full) {
#pragma unroll
        for (int v = 0; v < 8; ++v)
          p[(size_t)v * H2D] = fmaxf(acc[v] + bj, 0.0f);
      } else {
#pragma unroll
        for (int v = 0; v < 8; ++v) {
          int m = tile * 16 + v + 8 * hi;
          if (m < N) h2[(size_t)m * H2D + j] = fmaxf(acc[v] + bj, 0.0f);
        }
      }
    }
  }
}

// ---------------------------------------------------------------------------
// Conv2 pre: h2w = cs * (h2 @ conv2_w)   [N,24]x[24,24]
// ---------------------------------------------------------------------------
__global__ __launch_bounds__(256) void conv2_gemm_kernel(
    const float* __restrict__ h2, const float* __restrict__ w /*[24,24]*/,
    const float* __restrict__ cs, float* __restrict__ out, int N)
{
  const int lane = threadIdx.x & 31;
  const int wave = threadIdx.x >> 5;
  const int ntiles = (N + 15) >> 4;
  const int tile = blockIdx.x * 8 + wave;
  if (tile >= ntiles) return;
  const int n = lane & 15, hi = lane >> 4;
  const bool full = (tile * 16 + 16) <= N;
  int node = tile * 16 + n;
  int rnode = node < N ? node : N - 1;

  v2f afrag[6];
#pragma unroll
  for (int s = 0; s < 6; ++s) {
    int k = s * 4 + 2 * hi;
    afrag[s][0] = h2[(size_t)rnode * H2D + k];
    afrag[s][1] = h2[(size_t)rnode * H2D + k + 1];
  }
  float csr[8];
#pragma unroll
  for (int v = 0; v < 8; ++v) {
    int m = tile * 16 + v + 8 * hi;
    csr[v] = cs[m < N ? m : N - 1];
  }
#pragma unroll
  for (int j0 = 0; j0 < 32; j0 += 16) {
    int j = j0 + n;
    v8f acc = {};
#pragma unroll
    for (int s = 0; s < 6; ++s) {
      int k = s * 4 + 2 * hi;
      v2f b;
      b[0] = (j < G2D) ? w[k * G2D + j]       : 0.0f;
      b[1] = (j < G2D) ? w[(k + 1) * G2D + j] : 0.0f;
      acc = __builtin_amdgcn_wmma_f32_16x16x4_f32(false, afrag[s], false, b,
                                                  (short)0, acc, false, false);
    }
    if (j < G2D) {
      float* p = out + (size_t)(tile * 16 + 8 * hi) * G2D + j;
      if (full) {
#pragma unroll
        for (int v = 0; v < 8; ++v)
          p[(size_t)v * G2D] = acc[v] * csr[v];
      } else {
#pragma unroll
        for (int v = 0; v < 8; ++v) {
          int m = tile * 16 + v + 8 * hi;
          if (m < N) out[(size_t)m * G2D + j] = acc[v] * csr[v];
        }
      }
    }
  }
}

// ---------------------------------------------------------------------------
// Per-graph mean pooling (sums + counts); graph_ids sorted so contention low
// ---------------------------------------------------------------------------
__global__ __launch_bounds__(256) void pool_kernel(
    const float* __restrict__ agg2, const float* __restrict__ cd,
    const float* __restrict__ c2b, const int* __restrict__ gid,
    float* __restrict__ gsum, float* __restrict__ gcnt, int N)
{
  int idx = blockIdx.x * blockDim.x + threadIdx.x;
  int total = N * G2D;
  if (idx >= total) return;
  int node = idx / G2D;
  int c = idx - node * G2D;
  float v = agg2[idx] * cd[node] + c2b[c];
  int g = gid[node];
  atomicAdd(&gsum[g * G2D + c], v);
  if (c == 0) atomicAdd(&gcnt[g], 1.0f);
}

__global__ void final_kernel(
    const float* __restrict__ gsum, const float* __restrict__ gcnt,
    const float* __restrict__ w3, const float* __restrict__ b3,
    float* __restrict__ out, int G)
{
  int g = blockIdx.x * blockDim.x + threadIdx.x;
  if (g >= G) return;
  float inv = 1.0f / fmaxf(gcnt[g], 1.0f);
  float s = b3[0];
#pragma unroll
  for (int c = 0; c < G2D; ++c) s += gsum[g * G2D + c] * inv * w3[c];
  out[g] = s;
}

// ---------------------------------------------------------------------------
extern "C" void kernel_launch(void* const* d_in, const int* in_sizes, int n_in,
                              void* d_out, int out_size, void* d_ws, size_t ws_size,
                              hipStream_t stream) {
  const float* nd   = (const float*)d_in[0];
  const float* nr   = (const float*)d_in[1];
  const float* act  = (const float*)d_in[2];
  const int*   src  = (const int*)d_in[3];
  const int*   dst  = (const int*)d_in[4];
  const int*   gid  = (const int*)d_in[5];
  const float* fc1w = (const float*)d_in[6];
  const float* fc1b = (const float*)d_in[7];
  const float* bng  = (const float*)d_in[8];
  const float* bnb  = (const float*)d_in[9];
  const float* c1w  = (const float*)d_in[10];
  const float* c1b  = (const float*)d_in[11];
  const float* fc2w = (const float*)d_in[12];
  const float* fc2b = (const float*)d_in[13];
  const float* c2w  = (const float*)d_in[14];
  const float* c2b  = (const float*)d_in[15];
  const float* fc3w = (const float*)d_in[16];
  const float* fc3b = (const float*)d_in[17];

  const int N = in_sizes[0] / 8;
  const int E = in_sizes[3];
  const int G = out_size;
  const size_t n = (size_t)N;

  // Workspace layout (4-byte slots), with buffer reuse across pipeline stages.
  int*   odeg = (int*)d_ws;               // [N]
  int*   ideg = odeg + n;                 // [N]
  float* cs   = (float*)(ideg + n);       // [N]
  float* cd   = cs + n;                   // [N]
  float* bufA = cd + n;                   // [40N]  h1, later h2w
  float* bufB = bufA + 40 * n;            // [24N]  h1w, later h2
  float* bufC = bufB + 24 * n;            // [24N]  agg1, later agg2
  float* sums = bufC + 24 * n;            // [80]
  float* stat = sums + 2 * H1D;           // [80]
  float* gsum = stat + 2 * H1D;           // [G*24]
  float* gcnt = gsum + (size_t)G * G2D;   // [G]

  const int ntiles  = (N + 15) / 16;
  const int wblocks = (ntiles + 7) / 8;
  const unsigned totE = (unsigned)E * 24u;
  const int eblocks = (int)((totE + 255u) / 256u);

  // Zero all accumulators (graph-capture-legal memset nodes)
  hipMemsetAsync(odeg, 0, 2 * n * sizeof(int), stream);
  hipMemsetAsync(sums, 0, (size_t)(4 * H1D + G * (G2D + 1)) * sizeof(float), stream);
  hipMemsetAsync(bufC, 0, 24 * n * sizeof(float), stream);

  deg_count_kernel<<<(E + 255) / 256, 256, 0, stream>>>(src, dst, odeg, ideg, E);
  deg_norm_kernel<<<(N + 255) / 256, 256, 0, stream>>>(odeg, ideg, cs, cd, N);

  fc1_wmma_kernel<<<wblocks, 256, 0, stream>>>(nd, nr, fc1w, fc1b, bufA, N);
  bn_stats_kernel<<<2048, 256, 0, stream>>>(bufA, sums, N);
  bn_finalize_kernel<<<1, 64, 0, stream>>>(sums, bng, bnb, stat, 1.0f / (float)N);

  conv1_gemm_kernel<<<wblocks, 256, 0, stream>>>(bufA, stat, c1w, cs, bufB, N);
  scatter_kernel<<<eblocks, 256, 0, stream>>>(bufB, src, dst, bufC, totE);

  fc2_wmma_kernel<<<wblocks, 256, 0, stream>>>(bufC, cd, c1b, act, fc2w, fc2b, bufB, N);

  hipMemsetAsync(bufC, 0, 24 * n * sizeof(float), stream);  // re-zero for agg2
  conv2_gemm_kernel<<<wblocks, 256, 0, stream>>>(bufB, c2w, cs, bufA, N);
  scatter_kernel<<<eblocks, 256, 0, stream>>>(bufA, src, dst, bufC, totE);

  pool_kernel<<<(N * G2D + 255) / 256, 256, 0, stream>>>(bufC, cd, c2b, gid, gsum, gcnt, N);
  final_kernel<<<1, 64, 0, stream>>>(gsum, gcnt, fc3w, fc3b, (float*)d_out, G);
}